// IndexSVD_17772574671114
// MI455X (gfx1250) — compile-verified
//
#include <hip/hip_runtime.h>
#include <hip/hip_bf16.h>

// ---------------------------------------------------------------------------
// Problem constants (match reference setup_inputs()):
//   N=100000, D=256, D_MAJOR=64, DEG=32, B=64, EF1=256, EF2=64, steps=6, k=10
// ---------------------------------------------------------------------------
#define PN        100000
#define PD        256
#define PDMAJ     64
#define PDEG      32
#define PB        64
#define PEF1      256
#define PEF2      64
#define PSTEPS    6
#define VIS_WORDS ((PN + 31) / 32)      // 3125
#define BIGF      1e30f
#define CAND_N    2048                  // candidates handled per chunk

typedef __attribute__((ext_vector_type(2))) float        v2f;
typedef __attribute__((ext_vector_type(8))) float        v8f;
typedef __attribute__((ext_vector_type(4))) unsigned int v4u;
typedef __attribute__((ext_vector_type(4))) int          v4i;
typedef __attribute__((ext_vector_type(8))) int          v8i;

// ---------------------------------------------------------------------------
// Kernel 1: qp = query @ VT^T  via V_WMMA_F32_16X16X4_F32 (full f32 precision)
// ---------------------------------------------------------------------------
__global__ __launch_bounds__(32) void project_kernel(
    const float* __restrict__ Q,    // [64][256]
    const float* __restrict__ VT,   // [256][256]
    float* __restrict__ QP)         // [64][256]
{
    const int lane = threadIdx.x;
    const int h    = lane >> 4;
    const int r    = lane & 15;
    const int m0   = (blockIdx.x & 3) * 16;
    const int j0   = (blockIdx.x >> 2) * 16;

    const float* qrow = Q  + (size_t)(m0 + r) * PD;
    const float* vrow = VT + (size_t)(j0 + r) * PD;

    v8f acc = {};
    for (int kk = 0; kk < PD; kk += 4) {
        float4 q4 = *(const float4*)(qrow + kk);
        float4 v4 = *(const float4*)(vrow + kk);
        v2f a, b;
        a[0] = h ? q4.z : q4.x;  a[1] = h ? q4.w : q4.y;   // A: K = 2h, 2h+1
        b[0] = h ? v4.z : v4.x;  b[1] = h ? v4.w : v4.y;   // B: K = v + 2h
        acc = __builtin_amdgcn_wmma_f32_16x16x4_f32(
            false, a, false, b, (short)0, acc, false, false);
    }
#pragma unroll
    for (int v = 0; v < 8; ++v)
        QP[(size_t)(m0 + v + 8 * h) * PD + j0 + r] = acc[v];   // M = v + 8h
}

// ---------------------------------------------------------------------------
// Kernel 2: per-row squared norms under both metrics (one pass; also warms L2)
// ---------------------------------------------------------------------------
__global__ __launch_bounds__(256) void norms_kernel(
    const float* __restrict__ data, float* __restrict__ n64, float* __restrict__ n256)
{
    const int row  = blockIdx.x * 8 + (threadIdx.x >> 5);
    const int lane = threadIdx.x & 31;
    if (row >= PN) return;
    const float4* r4 = (const float4*)(data + (size_t)row * PD);
    float4 a = r4[lane];        // cols [4*lane, 4*lane+4)      (0..127)
    float4 b = r4[32 + lane];   // cols [128+4*lane, ...)       (128..255)
    float pa = a.x * a.x + a.y * a.y + a.z * a.z + a.w * a.w;
    float pb = b.x * b.x + b.y * b.y + b.z * b.z + b.w * b.w;
    float full  = pa + pb;
    float major = (lane < 16) ? pa : 0.0f;   // cols < 64 live in lanes 0..15 chunk a
    for (int off = 16; off; off >>= 1) {
        full  += __shfl_xor(full, off);
        major += __shfl_xor(major, off);
    }
    if (lane == 0) { n256[row] = full; n64[row] = major; }
}

// ---------------------------------------------------------------------------
// Helpers
// ---------------------------------------------------------------------------
__device__ __forceinline__ float l2dist(const float* __restrict__ row,
                                        const float* __restrict__ q, int dims)
{
    float acc = 0.0f;
    const float4* r4 = (const float4*)row;
#pragma unroll 4
    for (int t = 0; t < dims / 4; ++t) {
        float4 x = r4[t];
        float d0 = x.x - q[4 * t + 0], d1 = x.y - q[4 * t + 1];
        float d2 = x.z - q[4 * t + 2], d3 = x.w - q[4 * t + 3];
        acc = fmaf(d0, d0, acc); acc = fmaf(d1, d1, acc);
        acc = fmaf(d2, d2, acc); acc = fmaf(d3, d3, acc);
    }
    return acc;
}

__device__ void bitonic_sort(float* d, int* id, int n, int tid, int nt)
{
    for (int ks = 2; ks <= n; ks <<= 1) {
        for (int j = ks >> 1; j > 0; j >>= 1) {
            __syncthreads();
            for (int i = tid; i < n; i += nt) {
                int ix = i ^ j;
                if (ix > i) {
                    bool up = ((i & ks) == 0);
                    float di = d[i], dx = d[ix];
                    if ((di > dx) == up) {
                        d[i] = dx; d[ix] = di;
                        int t = id[i]; id[i] = id[ix]; id[ix] = t;
                    }
                }
            }
        }
    }
    __syncthreads();
}

// ---------------------------------------------------------------------------
// Beam-search kernel: one workgroup (8 wave32) per query. Dataset is
// L2-resident (102 MB < 192 MB), so candidate-row gathers are L2 hits.
// Distances run on the matrix pipe: per wave, 16 candidates form the A matrix
// of V_WMMA_F32_16X16X4_F32 with the query broadcast across B's columns;
// dist = ||x||^2 - 2*dot + ||q||^2.
// ---------------------------------------------------------------------------
template <int DIMS, int EF, int CHUNKS, int POOL2N>
__global__ __launch_bounds__(256) void beam_kernel(
    const float* __restrict__ data,     // [N][256]
    const int*   __restrict__ graph,    // [N][DEG]
    const float* __restrict__ qp,       // [B][256]
    const float* __restrict__ norms,    // [N] squared norms under this metric
    const int*   __restrict__ entries,  // entry ids, or nullptr => {0}
    int ecount,
    int*         __restrict__ beam_out, // stage1: [B][EF] ids
    int*         __restrict__ out_ids,  // stage2: [B][k]
    float*       __restrict__ out_d,    // stage2: [B][k]
    const int*   __restrict__ kptr)
{
    __shared__ unsigned visited[VIS_WORDS];
    __shared__ __align__(16) float qs[DIMS];
    __shared__ float qn_s;
    __shared__ float beam_d[EF];
    __shared__ int   beam_id[EF];
    __shared__ float cand_d[CAND_N];
    __shared__ int   cand_id[CAND_N];
    __shared__ float pool_d[POOL2N];
    __shared__ int   pool_id[POOL2N];

    const int tid  = threadIdx.x;
    const int nt   = blockDim.x;
    const int b    = blockIdx.x;
    const int lane = tid & 31;
    const int wave = tid >> 5;
    const int h    = lane >> 4;   // half-wave
    const int r    = lane & 15;   // row within WMMA tile

    // ---- init: visited bitset; query row -> LDS (TDM path when available) --
    for (int i = tid; i < VIS_WORDS; i += nt) visited[i] = 0u;
    if (tid == 0) qn_s = 0.0f;

#if __has_builtin(__builtin_amdgcn_tensor_load_to_lds) && __has_builtin(__builtin_amdgcn_s_wait_tensorcnt)
    if (tid < 32) {
        // Tensor DMA: 1 x DIMS f32 tile of the 1 x 256 row qp[b] -> LDS qs[].
        unsigned long long ga = (unsigned long long)(const void*)(qp + (size_t)b * PD);
        unsigned lds = (unsigned)(unsigned long long)(const void*)&qs[0]; // flat->LDS addr[31:0]
        v4u g0; v8i g1; v4i g2, g3; v8i g4;
        g0[0] = 1u;                                             // count=1 (user D#)
        g0[1] = lds;                                            // lds_addr
        g0[2] = (unsigned)(ga & 0xFFFFFFFFu);                   // global_addr lo
        g0[3] = (unsigned)((ga >> 32) & 0x01FFFFFFu) | (2u << 30); // addr hi | type=2
        g1[0] = (int)(2u << 16);          // wg_mask=0, data_size=2 (4 bytes)
        g1[1] = (int)(256u << 16);        // tensor_dim0[15:0]=256 in bits 63:48
        g1[2] = (int)(1u << 16);          // tensor_dim0 hi=0 | tensor_dim1[15:0]=1
        g1[3] = (int)((unsigned)DIMS << 16); // tensor_dim1 hi=0 | tile_dim0=DIMS
        g1[4] = 1;                        // tile_dim1=1, tile_dim2=0
        g1[5] = 256;                      // tensor_dim0_stride lo32 = 256
        g1[6] = 0;                        // stride hi | tensor_dim1_stride lo
        g1[7] = 0;
        g2 = (v4i){0, 0, 0, 0};
        g3 = (v4i){0, 0, 0, 0};
        g4 = (v8i){0, 0, 0, 0, 0, 0, 0, 0};
        __builtin_amdgcn_tensor_load_to_lds(g0, g1, g2, g3, g4, 0);
        __builtin_amdgcn_s_wait_tensorcnt(0);                   // s_wait_tensorcnt 0
    }
#else
    for (int i = tid; i < DIMS; i += nt) qs[i] = qp[(size_t)b * PD + i];
#endif
    __syncthreads();

    // ||q||^2 (block reduction)
    {
        float p = 0.0f;
        for (int i = tid; i < DIMS; i += nt) p = fmaf(qs[i], qs[i], p);
        atomicAdd(&qn_s, p);
    }
    __syncthreads();
    const float qn = qn_s;

    // ---- entry phase: distances of (<=256) entries, sorted, top-EF beam ----
    for (int i = tid; i < 256; i += nt) {
        if (i < ecount) {
            int id = entries ? entries[(size_t)b * ecount + i] : 0;
            cand_id[i] = id;
            atomicOr(&visited[id >> 5], 1u << (id & 31));
            cand_d[i] = l2dist(data + (size_t)id * PD, qs, DIMS);
        } else {
            cand_id[i] = 0;
            cand_d[i]  = BIGF;
        }
    }
    bitonic_sort(cand_d, cand_id, 256, tid, nt);
    for (int i = tid; i < EF; i += nt) { beam_d[i] = cand_d[i]; beam_id[i] = cand_id[i]; }
    __syncthreads();

    // ---- fixed-step best-first expansion ----------------------------------
    for (int step = 0; step < PSTEPS; ++step) {
        for (int i = tid; i < POOL2N; i += nt) {
            if (i < EF) { pool_d[i] = beam_d[i]; pool_id[i] = beam_id[i]; }
            else        { pool_d[i] = BIGF;      pool_id[i] = 0; }
        }
        __syncthreads();

        for (int c = 0; c < CHUNKS; ++c) {
            // Phase A: gather neighbor ids, freshness test, prefetch rows
            for (int j = tid; j < CAND_N; j += nt) {
                int g    = c * CAND_N + j;
                int id   = graph[(size_t)beam_id[g / PDEG] * PDEG + (g % PDEG)];
                cand_id[j] = id;
                unsigned w = visited[id >> 5];
                bool fresh = !((w >> (id & 31)) & 1u);
                cand_d[j]  = fresh ? -1.0f : BIGF;
                if (fresh)
                    __builtin_prefetch(data + (size_t)id * PD, 0, 0); // global_prefetch_b8
            }
            __syncthreads();
            // Phase B: mark visited
            for (int j = tid; j < CAND_N; j += nt)
                atomicOr(&visited[cand_id[j] >> 5], 1u << (cand_id[j] & 31));
            __syncthreads();

            // Phase C: WMMA distances. Each wave owns 16 candidates per sweep.
            for (int base = wave * 16; base < CAND_N; base += 16 * 8) {
                float flag = cand_d[base + r];              // <0 => fresh (both halves)
                int   id   = (flag < 0.0f) ? cand_id[base + r] : 0;  // stale -> row 0 (hot)
                float nrm  = norms[id];
                const float* row = data + (size_t)id * PD;
                v8f acc = {};
                for (int kk = 0; kk < DIMS; kk += 4) {
                    v2f a, bb;
                    float2 ld = *(const float2*)(row + kk + 2 * h);  // A: K = 2h, 2h+1
                    a[0] = ld.x; a[1] = ld.y;
                    float4 qf = *(const float4*)(qs + kk);           // B bcast: q[K]
                    bb[0] = h ? qf.z : qf.x;
                    bb[1] = h ? qf.w : qf.y;
                    acc = __builtin_amdgcn_wmma_f32_16x16x4_f32(
                        false, a, false, bb, (short)0, acc, false, false);
                }
                // D column 0: lane (16h) VGPR v holds dot(row v+8h, q).
#pragma unroll
                for (int v = 0; v < 8; ++v) {
                    float nv = __shfl(nrm,  8 * h + v);   // norm  of row v+8h
                    float fv = __shfl(flag, 8 * h + v);   // fresh of row v+8h
                    if (r == 0) {
                        float dist = fmaxf(nv - 2.0f * acc[v] + qn, 0.0f);
                        cand_d[base + 8 * h + v] = (fv < 0.0f) ? dist : BIGF;
                    }
                }
            }
            __syncthreads();

            // Partial selection: sort the 2048 chunk, keep its best EF
            bitonic_sort(cand_d, cand_id, CAND_N, tid, nt);
            for (int i = tid; i < EF; i += nt) {
                pool_d[EF + c * EF + i]  = cand_d[i];
                pool_id[EF + c * EF + i] = cand_id[i];
            }
            __syncthreads();
        }

        bitonic_sort(pool_d, pool_id, POOL2N, tid, nt);
        for (int i = tid; i < EF; i += nt) { beam_d[i] = pool_d[i]; beam_id[i] = pool_id[i]; }
        __syncthreads();
    }

    // ---- outputs ----------------------------------------------------------
    if (beam_out)
        for (int i = tid; i < EF; i += nt)
            beam_out[(size_t)b * EF + i] = beam_id[i];
    if (out_ids) {
        int k = *kptr;
        for (int i = tid; i < k; i += nt) {
            out_ids[(size_t)b * k + i] = beam_id[i];
            out_d[(size_t)b * k + i]   = beam_d[i];
        }
    }
}

// ---------------------------------------------------------------------------
// Host-side launcher (graph-capture safe)
// ---------------------------------------------------------------------------
extern "C" void kernel_launch(void* const* d_in, const int* in_sizes, int n_in,
                              void* d_out, int out_size, void* d_ws, size_t ws_size,
                              hipStream_t stream)
{
    const float* query = (const float*)d_in[0];
    const float* data  = (const float*)d_in[1];
    const float* VT    = (const float*)d_in[2];
    const int*   graph = (const int*)d_in[3];
    const int*   kptr  = (const int*)d_in[4];
    // d_in[5] = ef_search (=64), baked into template params

    float* qp   = (float*)d_ws;                       // [64][256]
    int*   s1   = (int*)(qp + (size_t)PB * PD);       // [64][256]
    float* n64  = (float*)(s1 + (size_t)PB * PEF1);   // [N]
    float* n256 = n64 + PN;                           // [N]

    // 1) qp = query @ VT^T  (f32 WMMA)
    project_kernel<<<dim3(PB / 16 * PD / 16), dim3(32), 0, stream>>>(query, VT, qp);

    // 2) per-row norms under both metrics (single pass; warms L2 with `data`)
    norms_kernel<<<dim3((PN + 7) / 8), dim3(256), 0, stream>>>(data, n64, n256);

    // 3) stage 1: 64-dim metric, ef=256, entry node 0, 4 chunks/step
    beam_kernel<PDMAJ, PEF1, 4, 2048><<<dim3(PB), dim3(256), 0, stream>>>(
        data, graph, qp, n64, nullptr, 1, s1, nullptr, nullptr, kptr);

    // 4) stage 2: full 256-dim metric, ef=64, 256 entries from stage 1
    int*   out_ids = (int*)d_out;
    float* out_d   = (float*)d_out + (size_t)PB * 10;
    beam_kernel<PD, PEF2, 1, 128><<<dim3(PB), dim3(256), 0, stream>>>(
        data, graph, qp, n256, s1, PEF1, nullptr, out_ids, out_d, kptr);
}